// VqVae_47519518163259
// MI455X (gfx1250) — compile-verified
//
#include <hip/hip_runtime.h>
#include <math.h>

typedef float v2f __attribute__((ext_vector_type(2)));
typedef float v8f __attribute__((ext_vector_type(8)));

#define EPSBN 1e-5f

#define WMMA4(A, B, C) \
    __builtin_amdgcn_wmma_f32_16x16x4_f32(false, (A), false, (B), (short)0, (C), false, false)

// ---------------------------------------------------------------------------
// Block-level partial reduction (sum, sumsq) for VALU kernels
// ---------------------------------------------------------------------------
__device__ __forceinline__ void tile_reduce_store(float v, float vsq,
                                                  float* pp, float* pq, int slot) {
    __shared__ float s1[256];
    __shared__ float s2[256];
    int t = threadIdx.x;
    s1[t] = v; s2[t] = vsq;
    __syncthreads();
    for (int s = blockDim.x >> 1; s > 0; s >>= 1) {
        if (t < s) { s1[t] += s1[t + s]; s2[t] += s2[t + s]; }
        __syncthreads();
    }
    if (t == 0) { pp[slot] = s1[0]; pq[slot] = s2[0]; }
}

// Finalize stats: 8 blocks (one per channel), fixed-order accumulation.
__global__ void k_reduce(const float* __restrict__ pp, const float* __restrict__ pq,
                         float* __restrict__ stat, int perChan) {
    __shared__ float s1[256];
    __shared__ float s2[256];
    int c = blockIdx.x, t = threadIdx.x;
    float s = 0.f, q = 0.f;
    for (int i = t; i < perChan; i += 256) { s += pp[c * perChan + i]; q += pq[c * perChan + i]; }
    s1[t] = s; s2[t] = q;
    __syncthreads();
    for (int k = 128; k > 0; k >>= 1) {
        if (t < k) { s1[t] += s1[t + k]; s2[t] += s2[t + k]; }
        __syncthreads();
    }
    if (t == 0) { stat[c] = s1[0]; stat[8 + c] = s2[0]; }
}

// ---------------------------------------------------------------------------
// Encoder conv1: [64,1,512,512] -> [64,8,256,256], 3x3 s2 p1, no bias, + stats
// Unconditional clamped loads, mask folded into the weight. grid 131072 x 256
// ---------------------------------------------------------------------------
__global__ void k_conv1(const float* __restrict__ x, const float* __restrict__ w,
                        float* __restrict__ out, float* pp, float* pq) {
    int b = blockIdx.x;
    int yo = b & 255, c = (b >> 8) & 7, n = b >> 11;
    int xo = threadIdx.x;
    const float* xin = x + (size_t)n * 262144;
    float acc = 0.f;
#pragma unroll
    for (int ky = 0; ky < 3; ++ky) {
        int iy = 2 * yo - 1 + ky;
        bool yok = (unsigned)iy < 512u;
        int row = (iy & 511) << 9;
#pragma unroll
        for (int kx = 0; kx < 3; ++kx) {
            int ix = 2 * xo - 1 + kx;
            bool ok = yok && ((unsigned)ix < 512u);
            float msk = ok ? 1.f : 0.f;
            acc += xin[row + (ix & 511)] * msk * w[c * 9 + ky * 3 + kx];
        }
    }
    out[((size_t)(n * 8 + c) << 16) + (yo << 8) + xo] = acc;
    tile_reduce_store(acc, acc * acc, pp, pq, c * 16384 + n * 256 + yo);
}

// BN (batch stats) + ReLU in-place over a [64,8,256,256] buffer. grid 131072x256.
__global__ void k_bnrelu(float* __restrict__ buf, const float* __restrict__ st,
                         const float* __restrict__ g, const float* __restrict__ bb,
                         float inv) {
    int i = blockIdx.x * 256 + threadIdx.x;
    int c = (i >> 16) & 7;
    float mean = st[c] * inv;
    float var  = st[8 + c] * inv - mean * mean;
    float sc   = g[c] * rsqrtf(var + EPSBN);
    float v = (buf[i] - mean) * sc + bb[c];
    buf[i] = fmaxf(v, 0.f);
}

// ---------------------------------------------------------------------------
// Encoder conv2 via WMMA implicit GEMM: [64,8,256,256] -> [64,8,128,128],
// 3x3 s2 p1. 9 taps x 2 K-steps = 18 WMMAs per 16-px tile. grid 8192 x 256.
// B operand padded to 16 out-channels in LDS (zeros) -> no guarded ds_loads.
// ---------------------------------------------------------------------------
__global__ void k_conv2_w(const float* __restrict__ in, const float* __restrict__ w,
                          float* __restrict__ out) {
    __shared__ float s_w[1152];                 // [16][8][9], top half zero
    int t = threadIdx.x;
    for (int i = t; i < 1152; i += 256) s_w[i] = (i < 576) ? w[i] : 0.f;
    __syncthreads();
    int lane = t & 31, wave = t >> 5, half = lane >> 4, m = lane & 15;
    int pbase = (blockIdx.x * 8 + wave) << 4;   // linear output pixel base
    int n  = pbase >> 14;
    int y  = (pbase >> 7) & 127;
    int x0 = pbase & 127;
    int k0 = half << 1;
    const float* xb = in + ((size_t)n << 19);   // 8*65536 floats per image
    const float* wrow = s_w + m * 72;           // [8][9] for out-channel m
    v8f acc = {};
#pragma unroll
    for (int ky = 0; ky < 3; ++ky) {
        int yy = 2 * y - 1 + ky;
        bool yok = (unsigned)yy < 256u;
        int rowoff = ((yy & 255) << 8) + (k0 << 16);
#pragma unroll
        for (int kx = 0; kx < 3; ++kx) {
            int xx = 2 * (x0 + m) - 1 + kx;
            bool ok = yok && ((unsigned)xx < 256u);
            float msk = ok ? 1.f : 0.f;
            int a = rowoff + (xx & 255);
            int tap = ky * 3 + kx;
            v2f a0, a1, b0, b1;
            a0.x = xb[a]             * msk;
            a0.y = xb[a + (1 << 16)] * msk;
            a1.x = xb[a + (4 << 16)] * msk;
            a1.y = xb[a + (5 << 16)] * msk;
            b0.x = wrow[(k0    ) * 9 + tap];
            b0.y = wrow[(k0 + 1) * 9 + tap];
            b1.x = wrow[(k0 + 4) * 9 + tap];
            b1.y = wrow[(k0 + 5) * 9 + tap];
            acc = WMMA4(a0, b0, acc);
            acc = WMMA4(a1, b1, acc);
        }
    }
#pragma unroll
    for (int r = 0; r < 8; ++r) {
        int pos = pbase + r + (half << 3);
        int pn = pos >> 14, prem = pos & 16383;
        if (m < 8) out[(pn << 17) + (m << 14) + prem] = acc[r];
    }
}

// ---------------------------------------------------------------------------
// ResBlock A via WMMA: t1 = conv3x3(relu(x)) + b1 ; deterministic stats(t1).
// grid 8192 x 256.
// ---------------------------------------------------------------------------
__global__ void k_rbA_w(const float* __restrict__ x, const float* __restrict__ w,
                        const float* __restrict__ bias, float* __restrict__ t1,
                        float* pp, float* pq) {
    __shared__ float s_w[1152];                 // [16][8][9], top half zero
    __shared__ float s_b[16];                   // padded bias
    __shared__ float s_sum[256], s_sq[256];
    int t = threadIdx.x;
    for (int i = t; i < 1152; i += 256) s_w[i] = (i < 576) ? w[i] : 0.f;
    if (t < 16) s_b[t] = (t < 8) ? bias[t] : 0.f;
    __syncthreads();
    int lane = t & 31, wave = t >> 5, half = lane >> 4, m = lane & 15;
    int pbase = (blockIdx.x * 8 + wave) << 4;
    int n  = pbase >> 14;
    int y  = (pbase >> 7) & 127;
    int x0 = pbase & 127;
    int k0 = half << 1;
    const float* xb = x + ((size_t)n << 17);
    const float* wrow = s_w + m * 72;
    v8f acc = {};
#pragma unroll
    for (int ky = 0; ky < 3; ++ky) {
        int yy = y - 1 + ky;
        bool yok = (unsigned)yy < 128u;
        int rowoff = ((yy & 127) << 7) + (k0 << 14);
#pragma unroll
        for (int kx = 0; kx < 3; ++kx) {
            int xx = x0 + m - 1 + kx;
            bool ok = yok && ((unsigned)xx < 128u);
            float msk = ok ? 1.f : 0.f;
            int a = rowoff + (xx & 127);
            int tap = ky * 3 + kx;
            v2f a0, a1, b0, b1;
            a0.x = fmaxf(xb[a],             0.f) * msk;
            a0.y = fmaxf(xb[a + (1 << 14)], 0.f) * msk;
            a1.x = fmaxf(xb[a + (4 << 14)], 0.f) * msk;
            a1.y = fmaxf(xb[a + (5 << 14)], 0.f) * msk;
            b0.x = wrow[(k0    ) * 9 + tap];
            b0.y = wrow[(k0 + 1) * 9 + tap];
            b1.x = wrow[(k0 + 4) * 9 + tap];
            b1.y = wrow[(k0 + 5) * 9 + tap];
            acc = WMMA4(a0, b0, acc);
            acc = WMMA4(a1, b1, acc);
        }
    }
    float bi = s_b[m];
    float ls = 0.f, lq = 0.f;
#pragma unroll
    for (int r = 0; r < 8; ++r) {
        int pos = pbase + r + (half << 3);
        int pn = pos >> 14, prem = pos & 16383;
        float d = acc[r] + bi;
        if (m < 8) t1[(pn << 17) + (m << 14) + prem] = d;
        ls += d; lq += d * d;
    }
    // deterministic per-channel block stats (lanes with m==c carry channel c)
    s_sum[t] = ls; s_sq[t] = lq;
    __syncthreads();
    if (t < 8) {
        float S = 0.f, Q = 0.f;
        for (int wv = 0; wv < 8; ++wv) {
            S += s_sum[wv * 32 + t] + s_sum[wv * 32 + t + 16];
            Q += s_sq [wv * 32 + t] + s_sq [wv * 32 + t + 16];
        }
        pp[t * 8192 + blockIdx.x] = S;
        pq[t * 8192 + blockIdx.x] = Q;
    }
}

// ---------------------------------------------------------------------------
// ResBlock B via WMMA: t2 = conv1x1(relu(bn(t1))) + b2 ; stats(t2).
// One 16x16x8 GEMM (2 WMMAs) per 16-pixel tile. grid 8192 x 256.
// ---------------------------------------------------------------------------
__global__ void k_rbB_w(const float* __restrict__ t1, const float* __restrict__ st,
                        const float* __restrict__ g, const float* __restrict__ be,
                        const float* __restrict__ w2, const float* __restrict__ b2,
                        float* __restrict__ t2, float* pp, float* pq, float inv) {
    __shared__ float sc[8], sh[8];
    __shared__ float s_w[128];                  // [16][8], top half zero
    __shared__ float s_b2[16];                  // padded bias
    __shared__ float s_sum[256], s_sq[256];
    int t = threadIdx.x;
    if (t < 8) {
        float mean = st[t] * inv;
        float var  = st[8 + t] * inv - mean * mean;
        float s = g[t] * rsqrtf(var + EPSBN);
        sc[t] = s; sh[t] = be[t] - mean * s;
    }
    if (t < 16) s_b2[t] = (t < 8) ? b2[t] : 0.f;
    if (t < 128) s_w[t] = (t < 64) ? w2[t] : 0.f;
    __syncthreads();
    int lane = t & 31, wave = t >> 5, half = lane >> 4, m = lane & 15;
    int pbase = (blockIdx.x * 8 + wave) << 4;
    int n = pbase >> 14, rem = pbase & 16383;
    const float* xb = t1 + ((size_t)n << 17) + rem;
    int k0 = half << 1;
    v2f a0, a1, b0, b1;
    a0.x = fmaxf(xb[((k0    ) << 14) + m] * sc[k0    ] + sh[k0    ], 0.f);
    a0.y = fmaxf(xb[((k0 + 1) << 14) + m] * sc[k0 + 1] + sh[k0 + 1], 0.f);
    a1.x = fmaxf(xb[((k0 + 4) << 14) + m] * sc[k0 + 4] + sh[k0 + 4], 0.f);
    a1.y = fmaxf(xb[((k0 + 5) << 14) + m] * sc[k0 + 5] + sh[k0 + 5], 0.f);
    b0.x = s_w[m * 8 + k0    ];
    b0.y = s_w[m * 8 + k0 + 1];
    b1.x = s_w[m * 8 + k0 + 4];
    b1.y = s_w[m * 8 + k0 + 5];
    v8f acc = {};
    acc = WMMA4(a0, b0, acc);
    acc = WMMA4(a1, b1, acc);
    float bi = s_b2[m];
    float ls = 0.f, lq = 0.f;
#pragma unroll
    for (int r = 0; r < 8; ++r) {
        int pos = pbase + r + (half << 3);
        int pn = pos >> 14, prem = pos & 16383;
        float d = acc[r] + bi;
        if (m < 8) t2[(pn << 17) + (m << 14) + prem] = d;
        ls += d; lq += d * d;
    }
    s_sum[t] = ls; s_sq[t] = lq;
    __syncthreads();
    if (t < 8) {
        float S = 0.f, Q = 0.f;
        for (int wv = 0; wv < 8; ++wv) {
            S += s_sum[wv * 32 + t] + s_sum[wv * 32 + t + 16];
            Q += s_sq [wv * 32 + t] + s_sq [wv * 32 + t + 16];
        }
        pp[t * 8192 + blockIdx.x] = S;
        pq[t * 8192 + blockIdx.x] = Q;
    }
}

// C: out = x + bn(t2).  grid 32768 x 256 over 8,388,608 elems
__global__ void k_rbC(const float* __restrict__ x, const float* __restrict__ t2,
                      const float* __restrict__ st, const float* __restrict__ g,
                      const float* __restrict__ be, float* __restrict__ out, float inv) {
    int i = blockIdx.x * 256 + threadIdx.x;
    int c = (i >> 14) & 7;
    float mean = st[c] * inv;
    float var  = st[8 + c] * inv - mean * mean;
    float sc   = g[c] * rsqrtf(var + EPSBN);
    out[i] = x[i] + (t2[i] - mean) * sc + be[c];
}

// ---------------------------------------------------------------------------
// Vector quantization via WMMA + branch-free packed-key argmin.
// grid 8192 x 256 (8 waves/block, one 16-position tile per wave).
// ---------------------------------------------------------------------------
__global__ void k_vq(const float* __restrict__ emb, const float* __restrict__ cb,
                     float* __restrict__ grad, float* __restrict__ idxout) {
    __shared__ float s_cb[128];
    __shared__ float s_cb2[16];
    int t = threadIdx.x;
    if (t < 128) s_cb[t] = cb[t];
    __syncthreads();
    if (t < 16) {
        float q = 0.f;
#pragma unroll
        for (int d = 0; d < 8; ++d) { float v = s_cb[t * 8 + d]; q += v * v; }
        s_cb2[t] = q;
    }
    __syncthreads();

    int lane = t & 31, wave = t >> 5, half = lane >> 4, m = lane & 15;
    int pbase = (blockIdx.x * 8 + wave) << 4;
    int p = pbase + m;
    int n = p >> 14, rem = p & 16383;
    int base = (n << 17) + rem;             // NCHW: ((n*8+c)<<14)+rem
    int k0 = half << 1;

    v2f a0, a1, b0, b1;
    a0.x = emb[base + ((k0    ) << 14)];
    a0.y = emb[base + ((k0 + 1) << 14)];
    a1.x = emb[base + ((k0 + 4) << 14)];
    a1.y = emb[base + ((k0 + 5) << 14)];
    b0.x = s_cb[m * 8 + k0];     b0.y = s_cb[m * 8 + k0 + 1];
    b1.x = s_cb[m * 8 + k0 + 4]; b1.y = s_cb[m * 8 + k0 + 5];

    v8f acc = {};
    acc = WMMA4(a0, b0, acc);
    acc = WMMA4(a1, b1, acc);

#pragma unroll
    for (int r = 0; r < 8; ++r) {
        float s = s_cb2[m] - 2.f * acc[r];          // |flat|^2 constant per row
        unsigned u = __float_as_uint(s);
        unsigned key = (u & 0x80000000u) ? ~u : (u | 0x80000000u);  // sortable
        unsigned long long k = ((unsigned long long)key << 32) | (unsigned)m;
#pragma unroll
        for (int off = 8; off >= 1; off >>= 1) {    // min over 16 codes, per half
            unsigned long long ko = __shfl_xor(k, off, 32);
            k = (ko < k) ? ko : k;                  // branch-free; ties -> low idx
        }
        int bi = (int)((unsigned)k & 15u);
        int pos = pbase + r + (half << 3);
        int pn = pos >> 14, prem = pos & 16383;
        if (m < 8) grad[(pn << 17) + (m << 14) + prem] = s_cb[bi * 8 + m];
        if (m == 0) idxout[pos] = (float)bi;
    }
}

// ---------------------------------------------------------------------------
// ConvTranspose 8->8, k3 s2 p1 op1 (gather): [64,8,128,128] -> [64,8,256,256]
// ReLU on read; masked unconditional loads; + stats. grid 131072 x 256.
// ---------------------------------------------------------------------------
__global__ void k_convt1(const float* __restrict__ in, const float* __restrict__ wt,
                         float* __restrict__ out, float* pp, float* pq) {
    int b = blockIdx.x;
    int oy = b & 255, o = (b >> 8) & 7, n = b >> 11;
    int ox = threadIdx.x;
    const float* xb = in + ((size_t)n << 17);
    float acc = 0.f;
#pragma unroll
    for (int ky = 0; ky < 3; ++ky) {
        int ty = oy + 1 - ky;
        int iy = ty >> 1;
        bool yok = !(ty & 1) && ((unsigned)iy < 128u);
        int row = (iy & 127) << 7;
#pragma unroll
        for (int kx = 0; kx < 3; ++kx) {
            int tx = ox + 1 - kx;
            int ix = tx >> 1;
            bool ok = yok && !(tx & 1) && ((unsigned)ix < 128u);
            float msk = ok ? 1.f : 0.f;
            int a = row + (ix & 127);
            for (int i = 0; i < 8; ++i) {
                float v = fmaxf(xb[(i << 14) + a], 0.f) * msk;
                acc += v * wt[(i * 8 + o) * 9 + ky * 3 + kx];
            }
        }
    }
    out[((size_t)(n * 8 + o) << 16) + (oy << 8) + ox] = acc;
    tile_reduce_store(acc, acc * acc, pp, pq, o * 16384 + n * 256 + oy);
}

// ConvTranspose 8->1 + tanh: [64,8,256,256] -> [64,1,512,512]. grid 32768 x 256.
__global__ void k_convt2(const float* __restrict__ in, const float* __restrict__ wt,
                         float* __restrict__ xhat) {
    int b = blockIdx.x;
    int oy = b & 511, n = b >> 9;
    const float* xb = in + ((size_t)n << 19);
    for (int rep = 0; rep < 2; ++rep) {
        int ox = threadIdx.x + rep * 256;
        float acc = 0.f;
#pragma unroll
        for (int ky = 0; ky < 3; ++ky) {
            int ty = oy + 1 - ky;
            int iy = ty >> 1;
            bool yok = !(ty & 1) && ((unsigned)iy < 256u);
            int row = (iy & 255) << 8;
#pragma unroll
            for (int kx = 0; kx < 3; ++kx) {
                int tx = ox + 1 - kx;
                int ix = tx >> 1;
                bool ok = yok && !(tx & 1) && ((unsigned)ix < 256u);
                float msk = ok ? 1.f : 0.f;
                int a = row + (ix & 255);
                for (int i = 0; i < 8; ++i)
                    acc += xb[(i << 16) + a] * msk * wt[i * 9 + ky * 3 + kx];
            }
        }
        xhat[(size_t)n * 262144 + (size_t)oy * 512 + ox] = tanhf(acc);
    }
}

// ---------------------------------------------------------------------------
// Host side
// ---------------------------------------------------------------------------
static void runRB(void* const* d_in, int base, const float* x, float* out,
                  float* t1, float* t2, float* pp, float* pq,
                  float* sA, float* sB, hipStream_t stream) {
    const float* w1  = (const float*)d_in[base + 0];
    const float* b1  = (const float*)d_in[base + 1];
    const float* g1  = (const float*)d_in[base + 2];
    const float* be1 = (const float*)d_in[base + 3];
    const float* w2  = (const float*)d_in[base + 4];
    const float* b2  = (const float*)d_in[base + 5];
    const float* g2  = (const float*)d_in[base + 6];
    const float* be2 = (const float*)d_in[base + 7];
    const float invRB = 1.f / 1048576.f;
    k_rbA_w<<<8192, 256, 0, stream>>>(x, w1, b1, t1, pp, pq);
    k_reduce<<<8, 256, 0, stream>>>(pp, pq, sA, 8192);
    k_rbB_w<<<8192, 256, 0, stream>>>(t1, sA, g1, be1, w2, b2, t2, pp, pq, invRB);
    k_reduce<<<8, 256, 0, stream>>>(pp, pq, sB, 8192);
    k_rbC<<<32768, 256, 0, stream>>>(x, t2, sB, g2, be2, out, invRB);
}

extern "C" void kernel_launch(void* const* d_in, const int* in_sizes, int n_in,
                              void* d_out, int out_size, void* d_ws, size_t ws_size,
                              hipStream_t stream) {
    (void)in_sizes; (void)n_in; (void)out_size; (void)ws_size;
    const float* x       = (const float*)d_in[0];
    const float* enc_w1  = (const float*)d_in[1];
    const float* enc_g1  = (const float*)d_in[2];
    const float* enc_b1  = (const float*)d_in[3];
    const float* enc_w2  = (const float*)d_in[4];
    // rb_e1: 5..12, rb_e2: 13..20
    const float* codebook= (const float*)d_in[21];
    // rb_d1: 22..29, rb_d2: 30..37
    const float* dec_wt1 = (const float*)d_in[38];
    const float* dec_g   = (const float*)d_in[39];
    const float* dec_b   = (const float*)d_in[40];
    const float* dec_wt2 = (const float*)d_in[41];

    // workspace layout (floats)
    float* ws    = (float*)d_ws;
    float* stats = ws;                  // 10 slots * 16 floats (use 256)
    float* pp    = ws + 256;            // 131072
    float* pq    = pp + 131072;         // 131072
    float* bufA  = pq + 131072;         // 33,554,432  (64x8x256x256)
    float* bufB  = bufA + 33554432;     // 8,388,608   (64x8x128x128)
    float* bufC  = bufB + 8388608;
    float* t1    = bufC + 8388608;
    float* t2    = t1 + 8388608;

    // output layout: xhat | emb | emb_grad | idx
    float* xhat = (float*)d_out;
    float* emb  = xhat + 16777216;
    float* grad = emb + 8388608;
    float* idxo = grad + 8388608;

    const float invBIG = 1.f / 4194304.f;

    // ---- encoder ----
    k_conv1<<<131072, 256, 0, stream>>>(x, enc_w1, bufA, pp, pq);
    k_reduce<<<8, 256, 0, stream>>>(pp, pq, stats + 0, 16384);
    k_bnrelu<<<131072, 256, 0, stream>>>(bufA, stats + 0, enc_g1, enc_b1, invBIG);
    k_conv2_w<<<8192, 256, 0, stream>>>(bufA, enc_w2, bufB);
    runRB(d_in, 5,  bufB, bufC, t1, t2, pp, pq, stats + 16, stats + 32, stream);   // rb_e1
    runRB(d_in, 13, bufC, emb,  t1, t2, pp, pq, stats + 48, stats + 64, stream);   // rb_e2 -> emb

    // ---- vector quantization (WMMA) ----
    k_vq<<<8192, 256, 0, stream>>>(emb, codebook, grad, idxo);

    // ---- decoder (reads quantized codes from emb_grad output slice) ----
    runRB(d_in, 22, grad, bufB, t1, t2, pp, pq, stats + 80,  stats + 96,  stream); // rb_d1
    runRB(d_in, 30, bufB, bufC, t1, t2, pp, pq, stats + 112, stats + 128, stream); // rb_d2
    k_convt1<<<131072, 256, 0, stream>>>(bufC, dec_wt1, bufA, pp, pq);
    k_reduce<<<8, 256, 0, stream>>>(pp, pq, stats + 144, 16384);
    k_bnrelu<<<131072, 256, 0, stream>>>(bufA, stats + 144, dec_g, dec_b, invBIG);
    k_convt2<<<32768, 256, 0, stream>>>(bufA, dec_wt2, xhat);
}